// HKLinear_67877663146207
// MI455X (gfx1250) — compile-verified
//
#include <hip/hip_runtime.h>
#include <hip/hip_bf16.h>

typedef float v2f __attribute__((ext_vector_type(2)));
typedef float v8f __attribute__((ext_vector_type(8)));
typedef int   v4i __attribute__((vector_size(4 * sizeof(int))));
typedef __attribute__((address_space(1))) v4i* gbl_v4i_ptr;
typedef __attribute__((address_space(3))) v4i* lds_v4i_ptr;

#define THRESHOLD 1e-4f
#define NCLUST 64
#define KT 32
#define LDS_STRIDE 36   // 32 + 4 pad: 144B rows -> float4 aligned, conflict-free frag reads
#define TILE_F (128 * LDS_STRIDE)

#if __has_builtin(__builtin_amdgcn_global_load_async_to_lds_b128) && \
    __has_builtin(__builtin_amdgcn_s_wait_asynccnt)
#define HAVE_ASYNC_LDS 1
#else
#define HAVE_ASYNC_LDS 0
#endif

// ---------------------------------------------------------------------------
// Kernel 0: zero the cluster-active flags (64 ints in d_ws)
// ---------------------------------------------------------------------------
__global__ void init_flags_kernel(int* __restrict__ flags) {
    if (threadIdx.x < NCLUST) flags[threadIdx.x] = 0;
}

// ---------------------------------------------------------------------------
// Kernel 1: routing.  4 rows per block, 64 clusters, softmax + threshold,
// OR-reduce per cluster into flags[].
// ---------------------------------------------------------------------------
#define RT_ROWS 4
__global__ __launch_bounds__(256)
void routing_kernel(const float* __restrict__ x,
                    const float* __restrict__ centroids,
                    int* __restrict__ flags,
                    int K) {
    __shared__ float xs[RT_ROWS * 1024];
    __shared__ float srow[RT_ROWS][NCLUST];
    __shared__ float smax[RT_ROWS];
    __shared__ float ssum[RT_ROWS];

    const int t = threadIdx.x;       // 0..255
    const int r = t >> 6;            // row in block: 0..3
    const int c = t & (NCLUST - 1);  // cluster id:  0..63

    const long rowBase = (long)blockIdx.x * RT_ROWS * K;
    for (int i = t; i < RT_ROWS * 1024 / 4; i += 256)
        ((float4*)xs)[i] = ((const float4*)(x + rowBase))[i];
    __syncthreads();

    float acc = 0.f;
    const float* cp = centroids + (long)c * K;
    const float* xr = xs + r * 1024;
    for (int k = 0; k < K; k += 4) {
        float4 cv = *(const float4*)(cp + k);
        acc += xr[k] * cv.x + xr[k + 1] * cv.y + xr[k + 2] * cv.z + xr[k + 3] * cv.w;
    }
    srow[r][c] = acc;
    __syncthreads();

    if (t < RT_ROWS) {
        float m = srow[t][0];
        for (int j = 1; j < NCLUST; ++j) m = fmaxf(m, srow[t][j]);
        float s = 0.f;
        for (int j = 0; j < NCLUST; ++j) s += __expf(srow[t][j] - m);
        smax[t] = m;
        ssum[t] = s;
    }
    __syncthreads();

    float p = __expf(acc - smax[r]) / ssum[r];
    if (p > THRESHOLD) atomicOr(&flags[c], 1);
}

// ---------------------------------------------------------------------------
// async copy helper: one 16B global -> LDS transfer per lane via the CDNA5
// async engine (tracked by ASYNCcnt, no staging VGPRs).
// builtin signature (from diagnostics): (v4i AS1*, v4i AS3*, imm offset, imm cpol)
// ---------------------------------------------------------------------------
#if HAVE_ASYNC_LDS
__device__ __forceinline__ void async_copy16(const float* g, float* l) {
    __builtin_amdgcn_global_load_async_to_lds_b128(
        (gbl_v4i_ptr)g, (lds_v4i_ptr)l, /*offset=*/0, /*cpol=*/0);
}
#endif

// per-thread share of one (A,B) tile fill: 4 passes x 2 matrices x 16B
__device__ __forceinline__ void fill_tile(const float* __restrict__ x,
                                          const float* __restrict__ w,
                                          float* la, float* lb,
                                          int m0, int n0, int K, int kt,
                                          int ldr, int ldc) {
#pragma unroll
    for (int p = 0; p < 4; ++p) {
        const int row = ldr + p * 32;
        const float* ga = x + (long)(m0 + row) * K + kt + ldc;
        const float* gb = w + (long)(n0 + row) * K + kt + ldc;
#if HAVE_ASYNC_LDS
        async_copy16(ga, la + row * LDS_STRIDE + ldc);
        async_copy16(gb, lb + row * LDS_STRIDE + ldc);
#else
        *(float4*)(la + row * LDS_STRIDE + ldc) = *(const float4*)ga;
        *(float4*)(lb + row * LDS_STRIDE + ldc) = *(const float4*)gb;
#endif
    }
}

// ---------------------------------------------------------------------------
// Kernel 2: main GEMM  out = mask(col) ? x @ W^T + bias : 0
// 128x128 block tile, 8 waves (2M x 4N), each wave 4x2 tiles of 16x16 f32,
// V_WMMA_F32_16X16X4_F32 inner product, double-buffered LDS fed by
// GLOBAL_LOAD_ASYNC_TO_LDS_B128.
// ---------------------------------------------------------------------------
__global__ __launch_bounds__(256)
void gemm_masked_kernel(const float* __restrict__ x,
                        const float* __restrict__ w,
                        const float* __restrict__ bias,
                        const int* __restrict__ assignment,
                        const int* __restrict__ flags,
                        float* __restrict__ out,
                        int M, int N, int K) {
    __shared__ float lds_a[2][TILE_F];
    __shared__ float lds_b[2][TILE_F];

    const int t     = threadIdx.x;
    const int wv    = t >> 5;      // wave 0..7
    const int l     = t & 31;      // lane
    const int lmod  = l & 15;
    const int lhalf = l >> 4;
    const int wm    = wv >> 2;     // 0..1 : wave M position (64 rows each)
    const int wn    = wv & 3;      // 0..3 : wave N position (32 cols each)
    const int m0    = blockIdx.y * 128;
    const int n0    = blockIdx.x * 128;

    v8f acc[4][2] = {};

    const int ldr = t >> 3;        // 0..31 row within pass
    const int ldc = (t & 7) * 4;   // float4 column start

    const int nk = K / KT;

    // prologue: start tile 0 into buffer 0
    fill_tile(x, w, lds_a[0], lds_b[0], m0, n0, K, 0, ldr, ldc);

    for (int i = 0; i < nk; ++i) {
        // my async writes for buf[i&1] done...
#if HAVE_ASYNC_LDS
        __builtin_amdgcn_s_wait_asynccnt(0);
#endif
        // ...then barrier: every wave's portion of buf[i&1] has landed, and
        // all waves finished reading buf[(i+1)&1] in iteration i-1.
        __syncthreads();

        if (i + 1 < nk)
            fill_tile(x, w, lds_a[(i + 1) & 1], lds_b[(i + 1) & 1],
                      m0, n0, K, (i + 1) * KT, ldr, ldc);

        const float* la = lds_a[i & 1];
        const float* lb = lds_b[i & 1];

#pragma unroll
        for (int kk = 0; kk < KT; kk += 4) {
            v2f afrag[4];
            v2f bfrag[2];
#pragma unroll
            for (int mt = 0; mt < 4; ++mt)
                afrag[mt] = *(const v2f*)&la[(wm * 64 + mt * 16 + lmod) * LDS_STRIDE
                                             + kk + lhalf * 2];
#pragma unroll
            for (int nt = 0; nt < 2; ++nt)
                bfrag[nt] = *(const v2f*)&lb[(wn * 32 + nt * 16 + lmod) * LDS_STRIDE
                                             + kk + lhalf * 2];
#pragma unroll
            for (int mt = 0; mt < 4; ++mt)
#pragma unroll
                for (int nt = 0; nt < 2; ++nt)
                    acc[mt][nt] = __builtin_amdgcn_wmma_f32_16x16x4_f32(
                        /*neg_a=*/false, afrag[mt],
                        /*neg_b=*/false, bfrag[nt],
                        /*c_mod=*/(short)0, acc[mt][nt],
                        /*reuse_a=*/false, /*reuse_b=*/false);
        }
    }

    // epilogue: bias + cluster mask (mask zeroes bias too, matching reference)
#pragma unroll
    for (int nt = 0; nt < 2; ++nt) {
        const int col = n0 + wn * 32 + nt * 16 + lmod;
        const int active = flags[assignment[col]];
        const float bv = bias[col];
#pragma unroll
        for (int mt = 0; mt < 4; ++mt) {
#pragma unroll
            for (int i = 0; i < 8; ++i) {
                const int row = m0 + wm * 64 + mt * 16 + i + lhalf * 8;
                const float v = active ? (acc[mt][nt][i] + bv) : 0.f;
                out[(long)row * N + col] = v;
            }
        }
    }
}

// ---------------------------------------------------------------------------
extern "C" void kernel_launch(void* const* d_in, const int* in_sizes, int n_in,
                              void* d_out, int out_size, void* d_ws, size_t ws_size,
                              hipStream_t stream) {
    const float* x      = (const float*)d_in[0];   // [4,2048,1024] f32
    const float* weight = (const float*)d_in[1];   // [4096,1024]  f32
    const float* bias   = (const float*)d_in[2];   // [4096]       f32
    const float* cent   = (const float*)d_in[3];   // [64,1024]    f32
    const int*   assign = (const int*)d_in[4];     // [4096]       i32
    float*       out    = (float*)d_out;           // [4,2048,4096] f32
    int*         flags  = (int*)d_ws;              // 64 ints scratch

    const int K = in_sizes[3] / NCLUST;            // 1024
    const int N = in_sizes[1] / K;                 // 4096
    const int M = in_sizes[0] / K;                 // 8192

    hipLaunchKernelGGL(init_flags_kernel, dim3(1), dim3(64), 0, stream, flags);
    hipLaunchKernelGGL(routing_kernel, dim3(M / RT_ROWS), dim3(256), 0, stream,
                       x, cent, flags, K);
    dim3 grid(N / 128, M / 128);
    hipLaunchKernelGGL(gemm_masked_kernel, grid, dim3(256), 0, stream,
                       x, weight, bias, assign, flags, out, M, N, K);
}